// NEURAL_PYSCF_WF_18451179503904
// MI455X (gfx1250) — compile-verified
//
#include <hip/hip_runtime.h>

typedef __attribute__((ext_vector_type(2))) float v2f;
typedef __attribute__((ext_vector_type(8))) float v8f;

#define B_DIM  128
#define NELEC  192
#define NAO    128
#define NMO    96
#define NCONF  32
#define LDET   80
#define MROWS  (B_DIM * NELEC)   // 24576
#define BS_STRIDE 130            // padded LDS stride for mo_w tile (bank spread)
#define S_STRIDE  84             // padded LDS stride for 80x80 det tile

// ---------------------------------------------------------------------------
// Kernel 1: mo[b,e,m] = sum_n ao[b,e,n] * mo_w[m,n]
// M = 24576, N = 96, K = 128 ; f32 WMMA 16x16x4.
// Each workgroup: 128 rows (8 waves x 16-row M-tile), all 6 N-tiles per wave.
// ---------------------------------------------------------------------------
__global__ __launch_bounds__(256) void mo_gemm_kernel(const float* __restrict__ ao,
                                                      const float* __restrict__ mo_w,
                                                      float* __restrict__ mo) {
  __shared__ float Bs[NMO * BS_STRIDE];   // 96 x 130 floats = 49,920 B

  const int tid = threadIdx.x;
  // Stage the entire mo_w (96x128) into LDS.
  for (int i = tid; i < NMO * NAO; i += 256) {
    int m = i >> 7;        // / 128
    int n = i & 127;
    Bs[m * BS_STRIDE + n] = mo_w[i];
  }
  __syncthreads();

  const int wave = tid >> 5;
  const int lane = tid & 31;
  const int lr   = lane & 15;   // low lane index (M for A frag, N for B frag)
  const int lh   = lane >> 4;   // half select (K pair 0/1 vs 2/3)
  const int rowbase = blockIdx.x * 128 + wave * 16;
  const int row     = rowbase + lr;

  v8f zero = {};
  v8f acc[6];
#pragma unroll
  for (int nt = 0; nt < 6; ++nt) acc[nt] = zero;

  const float* arow = ao + (size_t)row * NAO;

  for (int q = 0; q < 32; ++q) {
    const int kb2 = 4 * q + lh * 2;            // K base for this lane half
    v2f a = *(const v2f*)(arow + kb2);         // A[lr, kb2], A[lr, kb2+1] (8B aligned)
#pragma unroll
    for (int nt = 0; nt < 6; ++nt) {
      v2f b;
      b.x = Bs[(nt * 16 + lr) * BS_STRIDE + kb2];      // B[k, n] = mo_w[n, k]
      b.y = Bs[(nt * 16 + lr) * BS_STRIDE + kb2 + 1];
      acc[nt] = __builtin_amdgcn_wmma_f32_16x16x4_f32(
          false, a, false, b, (short)0, acc[nt], false, false);
    }
  }

  // C/D layout: VGPR v -> M = v + lh*8, N = nt*16 + lr
#pragma unroll
  for (int nt = 0; nt < 6; ++nt) {
    const int col = nt * 16 + lr;
#pragma unroll
    for (int v = 0; v < 8; ++v) {
      const int m = v + lh * 8;
      mo[(size_t)(rowbase + m) * NMO + col] = acc[nt][v];
    }
  }
}

// ---------------------------------------------------------------------------
// Kernel 2: per (b, conf) 80x80 gather + blocked LU (block 16) determinant.
// Trailing updates A22 -= L21 @ U12 done with f32 WMMA (A negated at load).
// One workgroup (256 threads / 8 waves) per determinant.
// ---------------------------------------------------------------------------
__global__ __launch_bounds__(256) void det_kernel(const float* __restrict__ mo,
                                                  const int* __restrict__ conf_idx,
                                                  float* __restrict__ dets) {
  __shared__ float S[LDET * S_STRIDE];   // 80 x 84 floats = 26,880 B
  __shared__ int   cidx[LDET];

  const int tid = threadIdx.x;
  const int c = blockIdx.x & (NCONF - 1);
  const int b = blockIdx.x >> 5;

  if (tid < LDET) cidx[tid] = conf_idx[c * LDET + tid];
  __syncthreads();

  // Gather: sub[i][j] = mo[b, conf_idx[c,i], conf_idx[c,j] >> 1]
  const float* mob = mo + (size_t)b * NELEC * NMO;
  for (int t = tid; t < LDET * LDET; t += 256) {
    int i = t / LDET;
    int j = t - i * LDET;
    S[i * S_STRIDE + j] = mob[cidx[i] * NMO + (cidx[j] >> 1)];
  }
  __syncthreads();

  const int r    = tid >> 4;    // 0..15 (row within diagonal block)
  const int cl   = tid & 15;    // 0..15 (col within diagonal block)
  const int wave = tid >> 5;
  const int lane = tid & 31;
  const int lr   = lane & 15;
  const int lh   = lane >> 4;

  for (int kb = 0; kb < LDET; kb += 16) {
    // --- 1) Factor 16x16 diagonal block in place (no pivoting) ---
    for (int t = 0; t < 16; ++t) {
      float m = 0.f;
      const bool act = (r > t);
      if (act) m = S[(kb + r) * S_STRIDE + kb + t] / S[(kb + t) * S_STRIDE + kb + t];
      __syncthreads();
      if (act) {
        if (cl == t)      S[(kb + r) * S_STRIDE + kb + t]  = m;   // store L multiplier
        else if (cl > t)  S[(kb + r) * S_STRIDE + kb + cl] -= m * S[(kb + t) * S_STRIDE + kb + cl];
      }
      __syncthreads();
    }

    const int rem = LDET - kb - 16;
    if (rem > 0) {
      // --- 2) Panel solves (disjoint regions -> run in parallel) ---
      if (tid < rem) {                      // U12 = L11^{-1} A12, one column/thread
        const int j = kb + 16 + tid;
        float x[16];
#pragma unroll
        for (int t = 0; t < 16; ++t) {
          float s = S[(kb + t) * S_STRIDE + j];
#pragma unroll
          for (int u = 0; u < 16; ++u)
            if (u < t) s -= S[(kb + t) * S_STRIDE + kb + u] * x[u];
          x[t] = s;
          S[(kb + t) * S_STRIDE + j] = s;
        }
      } else if (tid >= 128 && tid < 128 + rem) {   // L21 = A21 U11^{-1}, one row/thread
        const int i = kb + 16 + (tid - 128);
        float y[16];
#pragma unroll
        for (int t = 0; t < 16; ++t) {
          float s = S[i * S_STRIDE + kb + t];
#pragma unroll
          for (int u = 0; u < 16; ++u)
            if (u < t) s -= y[u] * S[(kb + u) * S_STRIDE + kb + t];
          y[t] = s / S[(kb + t) * S_STRIDE + kb + t];
          S[i * S_STRIDE + kb + t] = y[t];
        }
      }
      __syncthreads();

      // --- 3) Trailing update: A22 -= L21 @ U12 (K = 16) via WMMA ---
      const int nt = rem >> 4;     // tiles per dimension (4,3,2,1)
      for (int tile = wave; tile < nt * nt; tile += 8) {   // wave-uniform -> EXEC all 1s
        const int ti = tile / nt;
        const int tj = tile - ti * nt;
        const int r0 = kb + 16 + ti * 16;
        const int c0 = kb + 16 + tj * 16;

        v8f cacc;
#pragma unroll
        for (int v = 0; v < 8; ++v)
          cacc[v] = S[(r0 + v + lh * 8) * S_STRIDE + c0 + lr];

#pragma unroll
        for (int q = 0; q < 4; ++q) {
          const int k0 = kb + 4 * q + lh * 2;
          v2f a, bf;
          a.x  = -S[(r0 + lr) * S_STRIDE + k0];       // negate A: D = (-L)*U + C
          a.y  = -S[(r0 + lr) * S_STRIDE + k0 + 1];
          bf.x =  S[(k0    ) * S_STRIDE + c0 + lr];   // U12[k, n]
          bf.y =  S[(k0 + 1) * S_STRIDE + c0 + lr];
          cacc = __builtin_amdgcn_wmma_f32_16x16x4_f32(
              false, a, false, bf, (short)0, cacc, false, false);
        }

#pragma unroll
        for (int v = 0; v < 8; ++v)
          S[(r0 + v + lh * 8) * S_STRIDE + c0 + lr] = cacc[v];
      }
      __syncthreads();
    }
  }

  // det = product of U diagonal (f64 accumulate to dodge partial underflow)
  if (tid == 0) {
    double p = 1.0;
    for (int i = 0; i < LDET; ++i) p *= (double)S[i * S_STRIDE + i];
    dets[blockIdx.x] = (float)p;    // blockIdx.x == b*NCONF + c
  }
}

// ---------------------------------------------------------------------------
// Kernel 3: out[b] = sum_c dets[b,c] * ci_w[c]
// ---------------------------------------------------------------------------
__global__ __launch_bounds__(128) void ci_reduce_kernel(const float* __restrict__ dets,
                                                        const float* __restrict__ ci_w,
                                                        float* __restrict__ out) {
  const int b = threadIdx.x;
  if (b < B_DIM) {
    float s = 0.f;
    for (int c = 0; c < NCONF; ++c) s += dets[b * NCONF + c] * ci_w[c];
    out[b] = s;
  }
}

extern "C" void kernel_launch(void* const* d_in, const int* in_sizes, int n_in,
                              void* d_out, int out_size, void* d_ws, size_t ws_size,
                              hipStream_t stream) {
  const float* ao       = (const float*)d_in[0];   // (128,192,128) f32
  const float* mo_w     = (const float*)d_in[1];   // (96,128) f32
  const float* ci_w     = (const float*)d_in[2];   // (1,32) f32
  const int*   conf_idx = (const int*)d_in[3];     // (32,80) i32

  float* mo   = (float*)d_ws;                      // 24576*96 f32 = 9.44 MB
  float* dets = mo + (size_t)MROWS * NMO;          // 4096 f32

  mo_gemm_kernel<<<MROWS / 128, 256, 0, stream>>>(ao, mo_w, mo);
  det_kernel<<<B_DIM * NCONF, 256, 0, stream>>>(mo, conf_idx, dets);
  ci_reduce_kernel<<<1, 128, 0, stream>>>(dets, ci_w, (float*)d_out);
}